// GraphModel_56590489092382
// MI455X (gfx1250) — compile-verified
//
#include <hip/hip_runtime.h>
#include <hip/hip_bf16.h>
#include <math.h>

// Shapes from the reference
#define BB   8
#define NN   50000
#define EE   500000
#define EMB  8
#define CC   32

typedef __attribute__((ext_vector_type(2))) float v2f;
typedef __attribute__((ext_vector_type(8))) float v8f;

// ---------------------------------------------------------------------------
// h0[b,n,e] = x[b,n] * emb[n,e]   (vectorized: EMB = 8 = two float4)
// ---------------------------------------------------------------------------
__global__ void embed_kernel(const float* __restrict__ x,
                             const float* __restrict__ emb,
                             float* __restrict__ h0) {
    int idx = blockIdx.x * blockDim.x + threadIdx.x;   // over B*N
    if (idx >= BB * NN) return;
    int n = idx % NN;
    float xv = x[idx];
    const float4* er = (const float4*)(emb + (size_t)n * EMB);
    float4* hr = (float4*)(h0 + (size_t)idx * EMB);
    float4 e0 = er[0], e1 = er[1];
    float4 o0, o1;
    o0.x = xv * e0.x; o0.y = xv * e0.y; o0.z = xv * e0.z; o0.w = xv * e0.w;
    o1.x = xv * e1.x; o1.y = xv * e1.y; o1.z = xv * e1.z; o1.w = xv * e1.w;
    hr[0] = o0; hr[1] = o1;
}

// ---------------------------------------------------------------------------
// zero-fill
// ---------------------------------------------------------------------------
__global__ void fill_zero(float* __restrict__ p, size_t n) {
    size_t i = (size_t)blockIdx.x * blockDim.x + threadIdx.x;
    size_t stride = (size_t)gridDim.x * blockDim.x;
    for (; i < n; i += stride) p[i] = 0.0f;
}

// ---------------------------------------------------------------------------
// edge scatter: agg[b,dst,:] += w * h[b,src,:]
// One thread per (edge, batch): short dependency chains, max atomics in
// flight. float4 gathers (global_load_b128); L2-resident float atomics.
// ---------------------------------------------------------------------------
template <int C>
__global__ void edge_aggregate(const int* __restrict__ src,
                               const int* __restrict__ dst,
                               const float* __restrict__ w,
                               const float* __restrict__ h,
                               float* __restrict__ agg) {
    long long tid = (long long)blockIdx.x * blockDim.x + threadIdx.x;
    if (tid >= (long long)EE * BB) return;
    int e = (int)(tid % EE);
    int b = (int)(tid / EE);
    int s = src[e];
    int d = dst[e];
    float we = w[e];
    // prefetch the next block's source row for this batch (global_prefetch_b8)
    int e2 = e + blockDim.x;
    if (e2 < EE) {
        int s2 = src[e2];
        __builtin_prefetch(h + ((size_t)b * NN + s2) * C, 0, 1);
    }
    const float4* hr = (const float4*)(h + ((size_t)b * NN + s) * C);
    float* ar = agg + ((size_t)b * NN + d) * C;
#pragma unroll
    for (int c4 = 0; c4 < C / 4; ++c4) {
        float4 v = hr[c4];
        atomicAdd(ar + 4 * c4 + 0, we * v.x);
        atomicAdd(ar + 4 * c4 + 1, we * v.y);
        atomicAdd(ar + 4 * c4 + 2, we * v.z);
        atomicAdd(ar + 4 * c4 + 3, we * v.w);
    }
}

// ---------------------------------------------------------------------------
// Gated GEMM:  Y = relu( sigmoid((A@W + b) @ gw + gb) * (A@W + b) )
//   A: [M, K]  (M = B*N = 400000, divisible by 16)
//   W: [K, 32], bias: [32], gw: [32], gb: [1],  Y: [M, 32]
// One wave handles 16 rows x 32 cols as two 16x16 f32 WMMA accumulators,
// K accumulated in steps of 4 via V_WMMA_F32_16X16X4_F32 (fp32-exact).
// A-operand layout (16x4 f32, 2 VGPRs): lanes 0-15 -> K = {kk, kk+1},
// lanes 16-31 -> K = {kk+2, kk+3}; B-operand mirrors (rows across lanes).
// C/D layout: reg j: lanes 0-15 -> row j, lanes 16-31 -> row j+8.
// ---------------------------------------------------------------------------
template <int K>
__global__ void gemm_gate_wmma(const float* __restrict__ A,
                               const float* __restrict__ W,
                               const float* __restrict__ bias,
                               const float* __restrict__ gw,
                               const float* __restrict__ gb,
                               float* __restrict__ Y,
                               int M) {
    int wave = (blockIdx.x * blockDim.x + threadIdx.x) >> 5;
    int lane = threadIdx.x & 31;
    int m0 = wave * 16;
    if (m0 >= M) return;              // uniform per wave: EXEC stays all-ones
    int row = lane & 15;
    int hi = lane >> 4;               // 0: lanes 0-15, 1: lanes 16-31

    v8f c0 = {};                      // cols 0..15
    v8f c1 = {};                      // cols 16..31

#pragma unroll
    for (int kk = 0; kk < K; kk += 4) {
        int ka = kk + 2 * hi;
        v2f a, b0, b1;
        a.x = A[(size_t)(m0 + row) * K + ka + 0];
        a.y = A[(size_t)(m0 + row) * K + ka + 1];
        b0.x = W[(ka + 0) * CC + row];
        b0.y = W[(ka + 1) * CC + row];
        b1.x = W[(ka + 0) * CC + 16 + row];
        b1.y = W[(ka + 1) * CC + 16 + row];
        c0 = __builtin_amdgcn_wmma_f32_16x16x4_f32(false, a, false, b0,
                                                   (short)0, c0, false, false);
        c1 = __builtin_amdgcn_wmma_f32_16x16x4_f32(false, a, false, b1,
                                                   (short)0, c1, false, false);
    }

    float bc0 = bias[row];
    float bc1 = bias[16 + row];
    float gw0 = gw[row];
    float gw1 = gw[16 + row];
    float gbv = gb[0];

#pragma unroll
    for (int j = 0; j < 8; ++j) {
        float y0 = c0[j] + bc0;       // row (m0 + j + 8*hi), col row
        float y1 = c1[j] + bc1;       // row (m0 + j + 8*hi), col 16+row
        // per-row dot with gate weights; reduce across this 16-lane half-group
        float t = y0 * gw0 + y1 * gw1;
        t += __shfl_xor(t, 8, 32);
        t += __shfl_xor(t, 4, 32);
        t += __shfl_xor(t, 2, 32);
        t += __shfl_xor(t, 1, 32);
        float gate = 1.0f / (1.0f + __expf(-(t + gbv)));
        float r0 = gate * y0; r0 = r0 > 0.0f ? r0 : 0.0f;
        float r1 = gate * y1; r1 = r1 > 0.0f ? r1 : 0.0f;
        int node = m0 + j + 8 * hi;
        Y[(size_t)node * CC + row] = r0;
        Y[(size_t)node * CC + 16 + row] = r1;
    }
}

// ---------------------------------------------------------------------------
// out init: out[b,o] = logb[o]
// ---------------------------------------------------------------------------
__global__ void out_init(const float* __restrict__ logb, float* __restrict__ out) {
    int i = threadIdx.x;
    if (i < BB * 2) out[i] = logb[i & 1];
}

// ---------------------------------------------------------------------------
// out[b,o] += sum_i h[b, i] * logW[i, o]   (i over N*C = 1.6M)
// float4-vectorized grid-stride partials, wave reduce, lane-0 atomics.
// logW float4 pair {a,b} covers elements i0..i0+3:
//   a = {w(i0,0), w(i0,1), w(i0+1,0), w(i0+1,1)}
//   b = {w(i0+2,0), w(i0+2,1), w(i0+3,0), w(i0+3,1)}
// ---------------------------------------------------------------------------
__global__ void logits_kernel(const float* __restrict__ h,    // [B, N*C]
                              const float* __restrict__ logW, // [N*C, 2]
                              float* __restrict__ out) {
    const size_t total4 = (size_t)NN * CC / 4;   // 400000 groups of 4
    const float4* h4 = (const float4*)h;
    const float4* w4 = (const float4*)logW;
    size_t i = (size_t)blockIdx.x * blockDim.x + threadIdx.x;
    size_t stride = (size_t)gridDim.x * blockDim.x;
    float acc[BB][2];
#pragma unroll
    for (int b = 0; b < BB; ++b) { acc[b][0] = 0.0f; acc[b][1] = 0.0f; }
    for (; i < total4; i += stride) {
        float4 wa = w4[2 * i + 0];
        float4 wb = w4[2 * i + 1];
#pragma unroll
        for (int b = 0; b < BB; ++b) {
            float4 hv = h4[(size_t)b * total4 + i];
            acc[b][0] += hv.x * wa.x + hv.y * wa.z + hv.z * wb.x + hv.w * wb.z;
            acc[b][1] += hv.x * wa.y + hv.y * wa.w + hv.z * wb.y + hv.w * wb.w;
        }
    }
    int lane = threadIdx.x & 31;
#pragma unroll
    for (int b = 0; b < BB; ++b) {
#pragma unroll
        for (int o = 0; o < 2; ++o) {
            float v = acc[b][o];
            v += __shfl_xor(v, 16, 32);
            v += __shfl_xor(v, 8, 32);
            v += __shfl_xor(v, 4, 32);
            v += __shfl_xor(v, 2, 32);
            v += __shfl_xor(v, 1, 32);
            if (lane == 0) atomicAdd(out + b * 2 + o, v);
        }
    }
}

// ---------------------------------------------------------------------------
extern "C" void kernel_launch(void* const* d_in, const int* in_sizes, int n_in,
                              void* d_out, int out_size, void* d_ws, size_t ws_size,
                              hipStream_t stream) {
    const float* x        = (const float*)d_in[0];   // [B,N,1]
    const int*   edge_src = (const int*)  d_in[1];   // [E]
    const int*   edge_dst = (const int*)  d_in[2];   // [E]
    const float* edge_w   = (const float*)d_in[3];   // [E]
    const float* emb      = (const float*)d_in[4];   // [N,EMB]
    const float* W0       = (const float*)d_in[5];   // [EMB,C]
    const float* b0       = (const float*)d_in[6];   // [C]
    const float* gw0      = (const float*)d_in[7];   // [C,1]
    const float* gb0      = (const float*)d_in[8];   // [1]
    const float* W1       = (const float*)d_in[9];   // [C,C]
    const float* b1       = (const float*)d_in[10];  // [C]
    const float* gw1      = (const float*)d_in[11];  // [C,1]
    const float* gb1      = (const float*)d_in[12];  // [1]
    const float* logW     = (const float*)d_in[13];  // [N*C,2]
    const float* logb     = (const float*)d_in[14];  // [2]
    float* out = (float*)d_out;

    // workspace layout (floats)
    float* ws = (float*)d_ws;
    const size_t BN = (size_t)BB * NN;
    float* h0   = ws;                         // BN*EMB = 3.2M
    float* agg0 = h0 + BN * EMB;              // BN*EMB
    float* h1   = agg0 + BN * EMB;            // BN*CC  = 12.8M
    float* agg1 = h1 + BN * CC;               // BN*CC
    float* h2   = agg1 + BN * CC;             // BN*CC

    const int M = (int)BN;                    // 400000, divisible by 16
    const long long EB = (long long)EE * BB;  // 4M edge-batch pairs

    // 1) embed
    embed_kernel<<<(int)((BN + 255) / 256), 256, 0, stream>>>(x, emb, h0);

    // 2) layer 0: zero agg, scatter, gated GEMM (K=8)
    fill_zero<<<1024, 256, 0, stream>>>(agg0, BN * EMB);
    edge_aggregate<EMB><<<(int)((EB + 255) / 256), 256, 0, stream>>>(
        edge_src, edge_dst, edge_w, h0, agg0);
    // M/16 waves, 8 waves per 256-thread block
    gemm_gate_wmma<EMB><<<(M / 16 + 7) / 8, 256, 0, stream>>>(
        agg0, W0, b0, gw0, gb0, h1, M);

    // 3) layer 1: zero agg, scatter, gated GEMM (K=32)
    fill_zero<<<2048, 256, 0, stream>>>(agg1, BN * CC);
    edge_aggregate<CC><<<(int)((EB + 255) / 256), 256, 0, stream>>>(
        edge_src, edge_dst, edge_w, h1, agg1);
    gemm_gate_wmma<CC><<<(M / 16 + 7) / 8, 256, 0, stream>>>(
        agg1, W1, b1, gw1, gb1, h2, M);

    // 4) logits
    out_init<<<1, 32, 0, stream>>>(logb, out);
    logits_kernel<<<1024, 256, 0, stream>>>(h2, logW, out);
}